// EdgeEnhancedGCN_28398323761553
// MI455X (gfx1250) — compile-verified
//
#include <hip/hip_runtime.h>

typedef __attribute__((ext_vector_type(2))) float v2f;
typedef __attribute__((ext_vector_type(8))) float v8f;

#define BB 4
#define SS 256
#define FF 256
#define DD 64
#define OO 256
#define BS (BB * SS)                         // 1024
#define R_ELEMS ((size_t)BS * SS * OO)       // 67,108,864

// D = A(16x4,f32) * B(4x16,f32) + C  -- V_WMMA_F32_16X16X4_F32 (wave32)
__device__ __forceinline__ v8f wmma4(v2f a, v2f b, v8f c) {
  return __builtin_amdgcn_wmma_f32_16x16x4_f32(
      /*neg_a=*/false, a, /*neg_b=*/false, b,
      /*c_mod=*/(short)0, c, /*reuse_a=*/false, /*reuse_b=*/false);
}

// ---------------------------------------------------------------------------
// K1: rt[bi, o] = text[bi, :] @ W_attn[0:F, o] + b_attn[o]
// grid (OUT/16, BS/16) = (16, 64); one wave per 16x16 tile; K = 256.
// ---------------------------------------------------------------------------
__global__ __launch_bounds__(32) void k_rtext(const float* __restrict__ text,
                                              const float* __restrict__ Wattn,
                                              const float* __restrict__ battn,
                                              float* __restrict__ rt) {
  const int tn = blockIdx.x;
  const int tm = blockIdx.y;
  const int lane = threadIdx.x;
  const int lo = lane & 15;
  const int hi = lane >> 4;
  const int hi2 = hi * 2;

  const float* aBase = text + (size_t)(tm * 16 + lo) * FF + hi2;
  const float* bBase = Wattn + tn * 16 + lo;

  v8f c = {};
  for (int k = 0; k < FF; k += 4) {
    v2f a = *(const v2f*)(aBase + k);
    v2f b;
    const float* bp = bBase + (size_t)(k + hi2) * OO;
    b.x = bp[0];
    b.y = bp[OO];
    c = wmma4(a, b, c);
  }
  const float bb = battn[tn * 16 + lo];
#pragma unroll
  for (int v = 0; v < 8; ++v)
    rt[(size_t)(tm * 16 + v + hi * 8) * OO + tn * 16 + lo] = c[v] + bb;
}

// ---------------------------------------------------------------------------
// K2: per (b,i):  r[j, o] = dep[b,i,j,:] @ Wdep + rt[bi, o]   (256x64 @ 64x256)
//     p_logit[j] = sum_o r[j, o] + (adj==0 ? -1e30 : 0)
// grid (1024); 256 threads = 8 waves.
// Wdep staged once per block into LDS (64 KB), packed as K-pairs so each
// B fragment is a single ds_load_b64. Each wave owns TWO j-strips (w, w+8)
// so every B fragment feeds two WMMAs. r stores are non-temporal (write-once,
// 256 MB stream; keep dep_embed resident in the 192 MB L2 for kernel 3).
// ---------------------------------------------------------------------------
__global__ __launch_bounds__(256) void k_r_p(const float* __restrict__ dep,
                                             const float* __restrict__ Wdep,
                                             const float* __restrict__ rt,
                                             const int* __restrict__ adj,
                                             float* __restrict__ r_out,
                                             float* __restrict__ p_out) {
  // wlds[p*OO + n] = (Wdep[2p, n], Wdep[2p+1, n]),  p = 0..31  (64 KB)
  __shared__ v2f wlds[32 * OO];

  const int bi = blockIdx.x;
  const int t = threadIdx.x;

  {  // cooperative staging: thread t owns column n = t, loops over k.
    float* wf = (float*)wlds;
    const int n = t;
    for (int k = 0; k < DD; ++k)
      wf[((size_t)(k >> 1) * OO + n) * 2 + (k & 1)] = Wdep[(size_t)k * OO + n];
  }
  __syncthreads();

  const int wave = t >> 5;
  const int lane = t & 31;
  const int lo = lane & 15;
  const int hi = lane >> 4;
  const int hi2 = hi * 2;

  const float* depBI = dep + (size_t)bi * SS * DD;
  const float* rtBI  = rt + (size_t)bi * OO;
  float* rBI = r_out + (size_t)bi * SS * OO;

  const int j0a = wave * 16;          // strip A
  const int j0b = (wave + 8) * 16;    // strip B

  // Preload A fragments for both strips: 16(j) x 64(e) each, K-steps of 4.
  v2f afragA[16], afragB[16];
  const float* aA = depBI + (size_t)(j0a + lo) * DD + hi2;
  const float* aB = depBI + (size_t)(j0b + lo) * DD + hi2;
#pragma unroll
  for (int kk = 0; kk < 16; ++kk) {
    afragA[kk] = *(const v2f*)(aA + kk * 4);
    afragB[kk] = *(const v2f*)(aB + kk * 4);
  }

  float psA[8], psB[8];
#pragma unroll
  for (int v = 0; v < 8; ++v) { psA[v] = 0.0f; psB[v] = 0.0f; }

  for (int tn = 0; tn < 16; ++tn) {
    const int n0 = tn * 16;
    v2f bfrag[16];
#pragma unroll
    for (int kk = 0; kk < 16; ++kk)
      bfrag[kk] = wlds[(kk * 2 + hi) * OO + n0 + lo];

    v8f c0 = {}, c1 = {};
#pragma unroll
    for (int kk = 0; kk < 16; ++kk) {
      c0 = wmma4(afragA[kk], bfrag[kk], c0);
      c1 = wmma4(afragB[kk], bfrag[kk], c1);
    }

    const float rtv = rtBI[n0 + lo];
#pragma unroll
    for (int v = 0; v < 8; ++v) {
      const float v0 = c0[v] + rtv;
      const float v1 = c1[v] + rtv;
      __builtin_nontemporal_store(v0, &rBI[(size_t)(j0a + v + hi * 8) * OO + n0 + lo]);
      __builtin_nontemporal_store(v1, &rBI[(size_t)(j0b + v + hi * 8) * OO + n0 + lo]);
      psA[v] += v0;
      psB[v] += v1;
    }
  }

  // Row-sum reduction over the 16 lanes (same half-wave) that share a row.
#pragma unroll
  for (int v = 0; v < 8; ++v) {
    float sA = psA[v];
    float sB = psB[v];
    sA += __shfl_xor(sA, 1, 32);  sB += __shfl_xor(sB, 1, 32);
    sA += __shfl_xor(sA, 2, 32);  sB += __shfl_xor(sB, 2, 32);
    sA += __shfl_xor(sA, 4, 32);  sB += __shfl_xor(sB, 4, 32);
    sA += __shfl_xor(sA, 8, 32);  sB += __shfl_xor(sB, 8, 32);
    if (lo == 0) {
      const int ja = j0a + v + hi * 8;
      const int jb = j0b + v + hi * 8;
      const int ma = adj[(size_t)bi * SS + ja];
      const int mb = adj[(size_t)bi * SS + jb];
      p_out[(size_t)bi * SS + ja] = sA + (ma == 0 ? -1e30f : 0.0f);
      p_out[(size_t)bi * SS + jb] = sB + (mb == 0 ? -1e30f : 0.0f);
    }
  }
}

// ---------------------------------------------------------------------------
// K3: per (b,i): softmax over j (in place on p), then
//     pdep[bi, e] = sum_j p[j] * dep[b,i,j,e]
// grid (1024); 256 threads.
// ---------------------------------------------------------------------------
__global__ __launch_bounds__(256) void k_softmax_pdep(const float* __restrict__ dep,
                                                      float* __restrict__ p,
                                                      float* __restrict__ pdep) {
  const int bi = blockIdx.x;
  const int t = threadIdx.x;
  __shared__ float sp[SS];
  __shared__ float red[256];

  const float x = p[(size_t)bi * SS + t];
  red[t] = x;
  __syncthreads();
  for (int s = 128; s > 0; s >>= 1) {
    if (t < s) red[t] = fmaxf(red[t], red[t + s]);
    __syncthreads();
  }
  const float mx = red[0];
  __syncthreads();

  const float e = __expf(x - mx);
  red[t] = e;
  __syncthreads();
  for (int s = 128; s > 0; s >>= 1) {
    if (t < s) red[t] += red[t + s];
    __syncthreads();
  }
  const float pv = e / red[0];
  sp[t] = pv;
  p[(size_t)bi * SS + t] = pv;
  __syncthreads();

  // pdep: thread = (e_idx, group); 64 e-lanes x 4 j-groups, coalesced in e.
  const int e_idx = t & 63;
  const int g = t >> 6;
  const float* db = dep + (size_t)bi * SS * DD;
  float acc = 0.0f;
  for (int j = g; j < SS; j += 4)
    acc += sp[j] * db[(size_t)j * DD + e_idx];
  red[t] = acc;
  __syncthreads();
  if (g == 0)
    pdep[(size_t)bi * DD + e_idx] =
        red[e_idx] + red[e_idx + 64] + red[e_idx + 128] + red[e_idx + 192];
}

// ---------------------------------------------------------------------------
// K4: output_sum[bi, o] = text[bi, o] + pdep[bi, :] @ W_fc[:, o] + b_fc[o]
// grid (OUT/16, BS/16) = (16, 64); one wave per tile; K = 64. (F == OUT)
// ---------------------------------------------------------------------------
__global__ __launch_bounds__(32) void k_osum(const float* __restrict__ pdep,
                                             const float* __restrict__ Wfc,
                                             const float* __restrict__ bfc,
                                             const float* __restrict__ text,
                                             float* __restrict__ osum) {
  const int tn = blockIdx.x;
  const int tm = blockIdx.y;
  const int lane = threadIdx.x;
  const int lo = lane & 15;
  const int hi = lane >> 4;
  const int hi2 = hi * 2;

  const float* aBase = pdep + (size_t)(tm * 16 + lo) * DD + hi2;
  const float* bBase = Wfc + tn * 16 + lo;

  v8f c = {};
#pragma unroll
  for (int k = 0; k < DD; k += 4) {
    v2f a = *(const v2f*)(aBase + k);
    v2f b;
    const float* bp = bBase + (size_t)(k + hi2) * OO;
    b.x = bp[0];
    b.y = bp[OO];
    c = wmma4(a, b, c);
  }
  const float bb = bfc[tn * 16 + lo];
#pragma unroll
  for (int v = 0; v < 8; ++v) {
    const int m = tm * 16 + v + hi * 8;
    osum[(size_t)m * OO + tn * 16 + lo] = c[v] + bb + text[(size_t)m * FF + tn * 16 + lo];
  }
}

// ---------------------------------------------------------------------------
extern "C" void kernel_launch(void* const* d_in, const int* in_sizes, int n_in,
                              void* d_out, int out_size, void* d_ws, size_t ws_size,
                              hipStream_t stream) {
  (void)in_sizes; (void)n_in; (void)out_size; (void)ws_size;
  const float* text  = (const float*)d_in[0];
  const int*   adj   = (const int*)d_in[1];
  const float* dep   = (const float*)d_in[2];
  const float* Wattn = (const float*)d_in[3];
  const float* battn = (const float*)d_in[4];
  const float* Wfc   = (const float*)d_in[5];
  const float* bfc   = (const float*)d_in[6];

  float* out   = (float*)d_out;
  float* r_out = out;                                  // [B,S,S,OUT]
  float* osum  = out + R_ELEMS;                        // [B,S,OUT]
  float* p_out = osum + (size_t)BS * OO;               // [B,S,S]

  float* rt   = (float*)d_ws;                          // [BS, OUT]  (1 MB)
  float* pdep = rt + (size_t)BS * OO;                  // [BS, DEP]  (256 KB)

  hipLaunchKernelGGL(k_rtext, dim3(16, 64), dim3(32), 0, stream,
                     text, Wattn, battn, rt);
  hipLaunchKernelGGL(k_r_p, dim3(BS), dim3(256), 0, stream,
                     dep, Wattn + (size_t)FF * OO, rt, adj, r_out, p_out);
  hipLaunchKernelGGL(k_softmax_pdep, dim3(BS), dim3(256), 0, stream,
                     dep, p_out, pdep);
  hipLaunchKernelGGL(k_osum, dim3(16, 64), dim3(32), 0, stream,
                     pdep, Wfc, bfc, text, osum);
}